// MXFP4Embedding_50852412785248
// MI455X (gfx1250) — compile-verified
//
#include <hip/hip_runtime.h>
#include <stdint.h>

// MXFP4 (E2M1) block-scaled fake-quant of gathered embedding rows.
// Only the gathered rows are quantized (output-only validation): DRAM traffic
// ~64MiB read + 64MiB write -> ~5.5us at 23.3 TB/s. Data movement uses the
// CDNA5 async global->LDS path (ASYNCcnt), double-buffered, zero barriers
// (each thread consumes exactly the 16B it staged; per-wave s_wait_asynccnt).
// VALU is trimmed with a mantissa-truncation round (exact match to
// jnp.searchsorted ties-to-lower) and ds_swizzle_b32 block-max reduction.

#define FEAT 1024          // floats per embedding row (reference FEATURES)
#define TPB  256           // 8 waves of 32; one float4 per thread = one row
#define ROWS_PER_WG 8      // rows pipelined per workgroup

#if defined(__AMDGCN__)
#if __has_builtin(__builtin_amdgcn_global_load_async_to_lds_b128)
#define USE_ASYNC_BUILTIN 1
#endif
#if __has_builtin(__builtin_amdgcn_s_wait_asynccnt)
#define USE_WAIT_BUILTIN 1
#endif
#endif

// Builtin signature (probe-confirmed by hipcc diagnostic): param0 = v4i AS1
// (__device__), param1 = v4i AS3 (__shared__), then imm offset, imm cpol.
typedef int v4i_t __attribute__((vector_size(16)));
typedef __attribute__((address_space(1))) v4i_t g_v4i_t;
typedef __attribute__((address_space(3))) v4i_t l_v4i_t;

__device__ __forceinline__ void async_copy_b128(const float* gsrc, float* ldst) {
#if defined(__HIP_DEVICE_COMPILE__)
#if defined(USE_ASYNC_BUILTIN)
  __builtin_amdgcn_global_load_async_to_lds_b128(
      (g_v4i_t*)gsrc, (l_v4i_t*)ldst, /*offset=*/0, /*cpol=*/0);
#else
  uint32_t la = (uint32_t)(uintptr_t)(__attribute__((address_space(3))) void*)ldst;
  uint64_t ga = (uint64_t)(uintptr_t)gsrc;
  asm volatile("global_load_async_to_lds_b128 %0, %1, off"
               :: "v"(la), "v"(ga) : "memory");
#endif
#else
  (void)gsrc; (void)ldst;
#endif
}

template <int N>
__device__ __forceinline__ void wait_asynccnt() {
#if defined(__HIP_DEVICE_COMPILE__)
#if defined(USE_WAIT_BUILTIN)
  __builtin_amdgcn_s_wait_asynccnt(N);
#else
  asm volatile("s_wait_asynccnt %0" :: "i"(N) : "memory");
#endif
#endif
}

// max over the 8-lane group (xor 1,2,4) = block-of-32-floats max.
// ds_swizzle group-of-32 xor mode: offset = xor_mask<<10 | 0x1F (and_mask).
__device__ __forceinline__ float group8_max(float m) {
  m = fmaxf(m, __int_as_float(__builtin_amdgcn_ds_swizzle(__float_as_int(m), 0x041F)));
  m = fmaxf(m, __int_as_float(__builtin_amdgcn_ds_swizzle(__float_as_int(m), 0x081F)));
  m = fmaxf(m, __int_as_float(__builtin_amdgcn_ds_swizzle(__float_as_int(m), 0x101F)));
  return m;
}

// Round-to-nearest signed E2M1 on s = x/scale, ties to LOWER magnitude
// (jnp.searchsorted side='left'), then rescale. The grid above 1.0 is exactly
// "f32 truncated to 1 mantissa bit"; (bits + 0x1FFFFF) & 0xFFC00000 rounds to
// that grid with ties-down (carry into exponent handles 1.75/3.5/5 midpoints).
// E2M1's subnormal binade skips 0.75, so t<=0.75 is patched explicitly; the
// fmin(.,6) also maps t==inf (scale underflow) to 6 like the reference.
__device__ __forceinline__ float fp4_quant_dequant(float x, float scale) {
  const float t = fabsf(x) / scale;                 // IEEE divide == jnp
  const uint32_t u = __float_as_uint(t);
  const float r = __uint_as_float((u + 0x001FFFFFu) & 0xFFC00000u);
  const float qh = fmaxf(fminf(r, 6.0f), 1.0f);     // valid for t > 0.75
  const float ql = (t <= 0.25f) ? 0.0f : 0.5f;      // {0, 0.5} binade
  const float q  = (t <= 0.75f) ? ql : qh;
  return copysignf(q * scale, x);                    // sign*grid*scale
}

__global__ __launch_bounds__(TPB)
void mxfp4_embed_gather_kernel(const int* __restrict__ idx,
                               const float* __restrict__ emb,
                               float* __restrict__ out,
                               int nrows) {
  __shared__ float lds[2][FEAT];               // 8KB double buffer
  const int tid = threadIdx.x;
  const int rowBase = (int)blockIdx.x * ROWS_PER_WG;
  if (rowBase >= nrows) return;
  const int rem = nrows - rowBase;
  const int nloc = rem < ROWS_PER_WG ? rem : ROWS_PER_WG;

  // Prologue: stage row 0 of this workgroup's batch.
  {
    const int r0 = idx[rowBase];                               // uniform s_load
    async_copy_b128(emb + (size_t)r0 * FEAT + tid * 4, &lds[0][tid * 4]);
  }

  for (int j = 0; j < nloc; ++j) {
    if (j + 1 < nloc) {
      // Stage next row into the other buffer, then wait for current row.
      const int rn = idx[rowBase + j + 1];
      async_copy_b128(emb + (size_t)rn * FEAT + tid * 4,
                      &lds[(j + 1) & 1][tid * 4]);
      wait_asynccnt<1>();   // 2 outstanding -> oldest (row j) complete
    } else {
      wait_asynccnt<0>();   // drain: row j complete
    }

    // Each thread reads back exactly the 16B it staged -> no barrier needed.
    const float4 x = *(const float4*)&lds[j & 1][tid * 4];

    // Block max over 32 consecutive floats = 8 lanes x float4 (wave32).
    float m = fmaxf(fmaxf(fabsf(x.x), fabsf(x.y)),
                    fmaxf(fabsf(x.z), fabsf(x.w)));
    m = group8_max(m);
    const float scale = (m == 0.0f) ? 1.0f : (m / 6.0f);  // IEEE div by 6

    float4 o;
    o.x = fp4_quant_dequant(x.x, scale);
    o.y = fp4_quant_dequant(x.y, scale);
    o.z = fp4_quant_dequant(x.z, scale);
    o.w = fp4_quant_dequant(x.w, scale);

    *(float4*)(out + (size_t)(rowBase + j) * FEAT + tid * 4) = o;
  }
}

extern "C" void kernel_launch(void* const* d_in, const int* in_sizes, int n_in,
                              void* d_out, int out_size, void* d_ws, size_t ws_size,
                              hipStream_t stream) {
  (void)n_in; (void)d_ws; (void)ws_size; (void)out_size;
  const int*   idx = (const int*)d_in[0];     // (4,4096) int32 token ids
  const float* emb = (const float*)d_in[1];   // (50257,1024) fp32 table
  float*       out = (float*)d_out;           // (4,4096,1024) fp32

  const int nrows  = in_sizes[0];             // 16384 gathered rows
  const int blocks = (nrows + ROWS_PER_WG - 1) / ROWS_PER_WG;

  mxfp4_embed_gather_kernel<<<blocks, TPB, 0, stream>>>(idx, emb, out, nrows);
}